// MOLAttn_16003048145617
// MI455X (gfx1250) — compile-verified
//
#include <hip/hip_runtime.h>
#include <hip/hip_bf16.h>
#include <math.h>

typedef __attribute__((ext_vector_type(2))) float v2f;
typedef __attribute__((ext_vector_type(4))) float v4f;
typedef __attribute__((ext_vector_type(8))) float v8f;

#define ENC_DIM 512
#define ADIM    512
#define GDIM    1536        // 3*ADIM
#define ND      5
#define HID     128
#define B_SZ    32
#define T_SZ    2000
#define TCHUNK  80
#define NSPLIT  25          // 25*80 == 2000
#define Q_SZ    (ENC_DIM / 4)   // 128 float4 slots per row

__device__ __forceinline__ float sigmoidf_(float x) { return 1.0f / (1.0f + expf(-x)); }

// ---------------------------------------------------------------------------
// Kernel 1: GRU input GEMMs with f32 WMMA (full precision, matches reference).
//   gx = rnn_c (32x512) @ W_ih^T (512x1536)
//   gh = h0    (32x512) @ W_hh^T (512x1536)
// One wave per 16x16 output tile; K accumulated 4 at a time via
// V_WMMA_F32_16X16X4_F32.
// A 16x4 layout : lanes 0-15 -> M=lane, VGPR{0,1}=K{0,1}; lanes 16-31 -> K{2,3}
// B 4x16 layout : lanes 0-15 -> N=lane, VGPR{0,1}=K{0,1}; lanes 16-31 -> K{2,3}
// C/D 16x16     : VGPR r, lanes 0-15 -> (M=r,   N=lane)
//                         lanes16-31 -> (M=8+r, N=lane-16)
// ---------------------------------------------------------------------------
__global__ void gemm_gru(const float* __restrict__ rnn_c,
                         const float* __restrict__ h0,
                         const float* __restrict__ W_ih,
                         const float* __restrict__ W_hh,
                         float* __restrict__ gx,
                         float* __restrict__ gh) {
  const int ntile = blockIdx.x;            // 0..95
  const int mtile = blockIdx.y;            // 0..1
  const int which = blockIdx.z;            // 0 = ih, 1 = hh
  const float* __restrict__ X = which ? h0   : rnn_c;   // (32, 512)
  const float* __restrict__ W = which ? W_hh : W_ih;    // (1536, 512) row-major
  float* __restrict__ G       = which ? gh   : gx;      // (32, 1536)

  const int lane = threadIdx.x;            // 0..31
  const int half = lane >> 4;              // 0 or 1
  const int l15  = lane & 15;
  const int m    = mtile * 16 + l15;
  const int n    = ntile * 16 + l15;

  v8f acc = {};
  for (int k = 0; k < ENC_DIM; k += 4) {
    const int kk = k + half * 2;
    v2f a, b;
    a.x = X[m * ENC_DIM + kk];
    a.y = X[m * ENC_DIM + kk + 1];
    b.x = W[n * ENC_DIM + kk];       // B[k][n] = W[n][k]  (D = X @ W^T)
    b.y = W[n * ENC_DIM + kk + 1];
    acc = __builtin_amdgcn_wmma_f32_16x16x4_f32(
        /*neg_a=*/false, a, /*neg_b=*/false, b,
        /*c_mod=*/(short)0, acc, /*reuse_a=*/false, /*reuse_b=*/false);
  }

#pragma unroll
  for (int r = 0; r < 8; ++r) {
    const int mg = mtile * 16 + half * 8 + r;
    G[mg * GDIM + n] = acc[r];
  }
}

// ---------------------------------------------------------------------------
// Kernel 2: GRU gates -> h  (32 x 512)
// ---------------------------------------------------------------------------
__global__ void gru_gates(const float* __restrict__ gx,
                          const float* __restrict__ gh,
                          const float* __restrict__ b_ih,
                          const float* __restrict__ b_hh,
                          const float* __restrict__ h0,
                          float* __restrict__ h) {
  const int b = blockIdx.x;
  const int i = threadIdx.x;   // 0..511
  const float xr = gx[b * GDIM + i]            + b_ih[i];
  const float xz = gx[b * GDIM + ADIM + i]     + b_ih[ADIM + i];
  const float xn = gx[b * GDIM + 2 * ADIM + i] + b_ih[2 * ADIM + i];
  const float hr = gh[b * GDIM + i]            + b_hh[i];
  const float hz = gh[b * GDIM + ADIM + i]     + b_hh[ADIM + i];
  const float hn = gh[b * GDIM + 2 * ADIM + i] + b_hh[2 * ADIM + i];
  const float r  = sigmoidf_(xr + hr);
  const float z  = sigmoidf_(xz + hz);
  const float n  = tanhf(xn + r * hn);
  h[b * ADIM + i] = (1.0f - z) * n + z * h0[b * ADIM + i];
}

// ---------------------------------------------------------------------------
// Kernel 3 (single block): params = tanh(h @ W1^T + b1) @ W2^T + b2
// then means / scales / softmax weights per batch -> mix[b][16]
// ---------------------------------------------------------------------------
__global__ void mlp_mix(const float* __restrict__ h,
                        const float* __restrict__ W1, const float* __restrict__ b1,
                        const float* __restrict__ W2, const float* __restrict__ b2,
                        float* __restrict__ mix) {
  __shared__ float hid[B_SZ * HID];   // 16 KB
  __shared__ float par[B_SZ * 16];
  const int tid = threadIdx.x;        // 0..511

  for (int o = tid; o < B_SZ * HID; o += 512) {
    const int b = o >> 7, j = o & (HID - 1);
    float s = b1[j];
    const float* hr = h + b * ADIM;
    const float* wr = W1 + j * ADIM;
    for (int d = 0; d < ADIM; ++d) s += hr[d] * wr[d];
    hid[b * HID + j] = tanhf(s);
  }
  __syncthreads();

  for (int o = tid; o < B_SZ * 3 * ND; o += 512) {
    const int b = o / (3 * ND), p = o % (3 * ND);
    float s = b2[p];
    const float* hr = hid + b * HID;
    const float* wr = W2 + p * HID;
    for (int d = 0; d < HID; ++d) s += hr[d] * wr[d];
    par[b * 16 + p] = s;
  }
  __syncthreads();

  if (tid < B_SZ) {
    const int b = tid;
    float wv[ND];
    float mx = -1e30f;
#pragma unroll
    for (int j = 0; j < ND; ++j) {
      wv[j] = par[b * 16 + 2 * ND + j];
      mx = fmaxf(mx, wv[j]);
    }
    float den = 0.0f;
#pragma unroll
    for (int j = 0; j < ND; ++j) { wv[j] = expf(wv[j] - mx); den += wv[j]; }
    const float inv = 1.0f / den;
#pragma unroll
    for (int j = 0; j < ND; ++j) {
      mix[b * 16 + j]          = fminf(fmaxf(sigmoidf_(par[b * 16 + j]) * 0.2f, 0.0f), (float)T_SZ);
      mix[b * 16 + ND + j]     = expf(par[b * 16 + ND + j]);
      mix[b * 16 + 2 * ND + j] = wv[j] * inv;
    }
  }
}

// ---------------------------------------------------------------------------
// Kernel 4: mixture-of-logistics weights + streaming partial context.
// grid (NSPLIT, B); 256 threads. Each block handles 80 t's.
// Threads split into two parities (128 threads each); each parity covers the
// full 512-wide d with one float4 (b128) non-temporal load per row, streaming
// every other row. Rows with w==0 (length mask) skipped -> ~half HBM traffic.
// Partials are per-(split, parity): deterministic two-stage reduction.
// ---------------------------------------------------------------------------
__global__ void attn_stream(const float* __restrict__ enc_z,
                            const float* __restrict__ mix,
                            const int* __restrict__ lens,
                            float* __restrict__ rnn_w_out,
                            float* __restrict__ partial) {
  const int b   = blockIdx.y;
  const int sc  = blockIdx.x;
  const int t0  = sc * TCHUNK;
  const int tid = threadIdx.x;   // 0..255
  const int par = tid >> 7;      // 0 or 1 (row parity)
  const int q   = tid & (Q_SZ - 1); // float4 slot 0..127
  __shared__ float wsh[TCHUNK];

  if (tid < TCHUNK) {
    const int t = t0 + tid;
    float w = 0.0f;
    if (t < lens[b]) {
      float acc = 1e-7f;
      const float tf = (float)t;
#pragma unroll
      for (int j = 0; j < ND; ++j) {
        const float m  = mix[b * 16 + j];
        const float s  = mix[b * 16 + ND + j];
        const float wt = mix[b * 16 + 2 * ND + j];
        const float f = sigmoidf_((tf + 0.5f - m) / s) - sigmoidf_((tf - 0.5f - m) / s);
        acc += wt * f;
      }
      w = acc;
    }
    rnn_w_out[b * T_SZ + t] = w;
    wsh[tid] = w;
  }
  __syncthreads();

  v4f acc = {0.0f, 0.0f, 0.0f, 0.0f};
  const v4f* base = (const v4f*)(enc_z + ((size_t)(b * T_SZ + t0)) * ENC_DIM);
  for (int tt = par; tt < TCHUNK; tt += 2) {
    const float w = wsh[tt];            // uniform per wave -> no divergence
    if (w != 0.0f) {
      const v4f v = __builtin_nontemporal_load(base + (size_t)tt * Q_SZ + q);
      acc += w * v;                     // streamed once: NT keeps L2 clean
    }
  }
  v4f* p = (v4f*)(partial + ((size_t)(b * NSPLIT + sc) * 2 + par) * ENC_DIM);
  p[q] = acc;
}

// ---------------------------------------------------------------------------
// Kernel 5: reduce 2*NSPLIT partials -> ctx (32 x 512), float4-wide.
// ---------------------------------------------------------------------------
__global__ void reduce_ctx(const float* __restrict__ partial, float* __restrict__ ctx) {
  const int b = blockIdx.x;
  const int q = threadIdx.x;   // 0..127
  v4f s = {0.0f, 0.0f, 0.0f, 0.0f};
  const v4f* p = (const v4f*)(partial + (size_t)b * NSPLIT * 2 * ENC_DIM);
  for (int k = 0; k < NSPLIT * 2; ++k)
    s += p[(size_t)k * Q_SZ + q];
  ((v4f*)(ctx + (size_t)b * ENC_DIM))[q] = s;
}

extern "C" void kernel_launch(void* const* d_in, const int* in_sizes, int n_in,
                              void* d_out, int out_size, void* d_ws, size_t ws_size,
                              hipStream_t stream) {
  (void)in_sizes; (void)n_in; (void)out_size; (void)ws_size;
  const float* enc_z = (const float*)d_in[0];
  const float* rnn_c = (const float*)d_in[1];
  const float* h0    = (const float*)d_in[2];
  const float* W_ih  = (const float*)d_in[3];
  const float* W_hh  = (const float*)d_in[4];
  const float* b_ih  = (const float*)d_in[5];
  const float* b_hh  = (const float*)d_in[6];
  const float* W1    = (const float*)d_in[7];
  const float* b1    = (const float*)d_in[8];
  const float* W2    = (const float*)d_in[9];
  const float* b2    = (const float*)d_in[10];
  const int*   lens  = (const int*)d_in[11];

  float* out   = (float*)d_out;
  float* ctx   = out;                       // 32*512  (first tuple element)
  float* rnn_w = out + B_SZ * ENC_DIM;      // 32*2000 (second tuple element)

  float* ws      = (float*)d_ws;
  float* gx      = ws;                      // 32*1536
  float* gh      = gx + B_SZ * GDIM;        // 32*1536
  float* h       = gh + B_SZ * GDIM;        // 32*512
  float* mix     = h + B_SZ * ADIM;         // 32*16
  float* partial = mix + B_SZ * 16;         // 32*25*2*512 (float4-aligned offset)

  hipLaunchKernelGGL(gemm_gru,    dim3(GDIM / 16, 2, 2), dim3(32),   0, stream,
                     rnn_c, h0, W_ih, W_hh, gx, gh);
  hipLaunchKernelGGL(gru_gates,   dim3(B_SZ),            dim3(ADIM), 0, stream,
                     gx, gh, b_ih, b_hh, h0, h);
  hipLaunchKernelGGL(mlp_mix,     dim3(1),               dim3(512),  0, stream,
                     h, W1, b1, W2, b2, mix);
  hipLaunchKernelGGL(attn_stream, dim3(NSPLIT, B_SZ),    dim3(256),  0, stream,
                     enc_z, mix, lens, rnn_w, partial);
  hipLaunchKernelGGL(reduce_ctx,  dim3(B_SZ),            dim3(Q_SZ), 0, stream,
                     partial, ctx);
}